// Dilation2dSingle_24120536334520
// MI455X (gfx1250) — compile-verified
//
#include <hip/hip_runtime.h>
#include <hip/hip_bf16.h>

#define NEGV (-10000.0f)

constexpr int TILE = 64;          // output tile edge
constexpr int RAD  = 5;           // halo radius (width=5 -> 11 taps)
constexpr int NTAP = 11;
constexpr int TW   = TILE + 2 * RAD;   // 74 : haloed tile edge
constexpr int SA   = 78;               // LDS stride for input tile (bank-conflict free)
constexpr int SB   = 66;               // LDS stride for horizontal-pass result
constexpr int NTHREADS = 256;          // 8 waves on wave32

#if defined(__gfx1250__) && __has_builtin(__builtin_amdgcn_global_load_async_to_lds_b32)
#define USE_ASYNC_LDS 1
#else
#define USE_ASYNC_LDS 0
#endif

#if USE_ASYNC_LDS
typedef __attribute__((address_space(1))) int  as1_int;   // global
typedef __attribute__((address_space(3))) int  as3_int;   // LDS
#endif

__global__ __launch_bounds__(NTHREADS)
void dilate2d_fused(const float* __restrict__ im,
                    const float* __restrict__ se_coef,
                    const float* __restrict__ se,
                    float* __restrict__ out,
                    int H, int W)
{
    __shared__ float A [TW * SA];   // 74*78*4 = 23088 B : haloed input tile
    __shared__ float Bs[TW * SB];   // 74*66*4 = 19536 B : after horizontal pass

    const int t   = threadIdx.x;
    const int tx0 = blockIdx.x * TILE;
    const int ty0 = blockIdx.y * TILE;
    const size_t planeOfs = (size_t)blockIdx.z * (size_t)H * (size_t)W;
    const float* plane  = im  + planeOfs;
    float*       oplane = out + planeOfs;

    // bias[k] = se_coef * se[k]  (uniform, cached)
    float bias[NTAP];
    {
        const float sc = se_coef[0];
        #pragma unroll
        for (int k = 0; k < NTAP; ++k) bias[k] = sc * se[k];
    }

    // ---------------- load haloed tile into LDS ----------------
    // In-range lanes: async DMA global->LDS (CDNA5 path). OOB lanes: NEG fill.
    // Address sets are disjoint, so no ordering hazard between the two.
    for (int idx = t; idx < TW * TW; idx += NTHREADS) {
        const int r  = idx / TW;
        const int c  = idx - r * TW;
        const int gy = ty0 + r - RAD;
        const int gx = tx0 + c - RAD;
        const bool inb = ((unsigned)gy < (unsigned)H) && ((unsigned)gx < (unsigned)W);
        if (inb) {
#if USE_ASYNC_LDS
            __builtin_amdgcn_global_load_async_to_lds_b32(
                (as1_int*)(plane + (size_t)gy * W + gx),
                (as3_int*)&A[r * SA + c],
                /*offset=*/0, /*cpol=*/0);
#else
            A[r * SA + c] = plane[(size_t)gy * W + gx];
#endif
        } else {
            A[r * SA + c] = NEGV;
        }
    }
#if USE_ASYNC_LDS
#if __has_builtin(__builtin_amdgcn_s_wait_asynccnt)
    __builtin_amdgcn_s_wait_asynccnt(0);
#else
    asm volatile("s_wait_asynccnt 0" ::: "memory");
#endif
#endif
    __syncthreads();

    // ---------------- horizontal pass (along W) ----------------
    // task = (row r in [0,74), 16-wide column strip); 11-deep register ring
    // reuses 10/11 window values per step: 26 LDS reads per 16 outputs.
    for (int task = t; task < TW * (TILE / 16); task += NTHREADS) {
        const int r  = task >> 2;          // 0..73
        const int c0 = (task & 3) << 4;    // 0,16,32,48
        const float* Ar = &A[r * SA + c0];
        float win[NTAP];
        #pragma unroll
        for (int k = 0; k < NTAP - 1; ++k) win[k] = Ar[k];
        #pragma unroll
        for (int i = 0; i < 16; ++i) {
            win[(i + NTAP - 1) % NTAP] = Ar[i + NTAP - 1];
            float m = win[i % NTAP] - bias[0];
            #pragma unroll
            for (int k = 1; k < NTAP; ++k)
                m = fmaxf(m, win[(i + k) % NTAP] - bias[k]);
            Bs[r * SB + c0 + i] = m;
        }
    }
    __syncthreads();

    // ---------------- vertical pass (along H) + store ----------------
    // thread t owns column (t&63), rows [rb, rb+16); same register ring.
    {
        const int c  = t & (TILE - 1);
        const int rb = (t >> 6) * (TILE / 4);   // 0,16,32,48
        float win[NTAP];
        #pragma unroll
        for (int k = 0; k < NTAP - 1; ++k) win[k] = Bs[(rb + k) * SB + c];
        #pragma unroll
        for (int i = 0; i < 16; ++i) {
            win[(i + NTAP - 1) % NTAP] = Bs[(rb + i + NTAP - 1) * SB + c];
            float m = win[i % NTAP] - bias[0];
            #pragma unroll
            for (int k = 1; k < NTAP; ++k)
                m = fmaxf(m, win[(i + k) % NTAP] - bias[k]);
            oplane[(size_t)(ty0 + rb + i) * W + (tx0 + c)] = m;  // coalesced
        }
    }
}

extern "C" void kernel_launch(void* const* d_in, const int* in_sizes, int n_in,
                              void* d_out, int out_size, void* d_ws, size_t ws_size,
                              hipStream_t stream)
{
    const float* im      = (const float*)d_in[0];   // [B,C,512,512] fp32
    const float* se_coef = (const float*)d_in[1];   // scalar
    const float* se      = (const float*)d_in[2];   // [11,1]
    float* out = (float*)d_out;

    const int H = 512, W = 512;
    const int planes = in_sizes[0] / (H * W);       // B*C = 256

    dim3 grid(W / TILE, H / TILE, planes);          // 8 x 8 x 256
    dilate2d_fused<<<grid, NTHREADS, 0, stream>>>(im, se_coef, se, out, H, W);
}